// SeedFormer_newup_83932250898498
// MI455X (gfx1250) — compile-verified
//
#include <hip/hip_runtime.h>
#include <math.h>

#define B_SZ  8
#define N_PTS 2048
#define KNN   20
#define CIN   256
#define COUT  128
#define DIM   64
#define PHID  64
#define HID   256
#define UPF   2
#define NUP   (N_PTS*UPF)
#define EPSB  1e-5f

typedef __attribute__((ext_vector_type(2))) float v2f;
typedef __attribute__((ext_vector_type(8))) float v8f;

static __device__ inline v8f zero8() {
  v8f z;
#pragma unroll
  for (int i = 0; i < 8; ++i) z[i] = 0.f;
  return z;
}

static __device__ inline v8f wmma_f32(v2f a, v2f b, v8f c) {
  // D = A(16x4 f32) * B(4x16 f32) + C(16x16 f32)
  return __builtin_amdgcn_wmma_f32_16x16x4_f32(false, a, false, b, (short)0, c, false, false);
}

// ---------------------------------------------------------------------------
// K1: h1 = relu(mv_w1 . concat(key,query) + mv_b1)   (B,256,N)
// grid(1024,4) block(128): wave -> one 16(ch) x 16(pt) tile
// ---------------------------------------------------------------------------
__global__ void k_mlp_hidden(const float* __restrict__ kf, const float* __restrict__ qf,
                             const float* __restrict__ w1, const float* __restrict__ b1,
                             float* __restrict__ h1) {
  const int lane = threadIdx.x & 31, hh = lane >> 4, lm = lane & 15;
  const int wvid = threadIdx.x >> 5;
  const int p0 = blockIdx.x * 16;
  const int b  = p0 / N_PTS, n0 = p0 % N_PTS;
  const int m0 = (blockIdx.y * 4 + wvid) * 16;
  v8f acc = zero8();
  for (int c0 = 0; c0 < 2 * CIN; c0 += 4) {
    const float* ap = w1 + (size_t)(m0 + lm) * (2 * CIN) + c0 + 2 * hh;
    v2f a; a.x = ap[0]; a.y = ap[1];
    const float* X = (c0 < CIN) ? (kf + ((size_t)b * CIN + c0) * N_PTS)
                                : (qf + ((size_t)b * CIN + (c0 - CIN)) * N_PTS);
    v2f bb;
    bb.x = X[(size_t)(2 * hh)     * N_PTS + n0 + lm];
    bb.y = X[(size_t)(2 * hh + 1) * N_PTS + n0 + lm];
    acc = wmma_f32(a, bb, acc);
  }
#pragma unroll
  for (int v = 0; v < 8; ++v) {
    int m = m0 + v + 8 * hh;
    float r = acc[v] + b1[m];
    h1[((size_t)b * CIN + m) * N_PTS + n0 + lm] = fmaxf(r, 0.f);
  }
}

// ---------------------------------------------------------------------------
// K2: value = mv_w2.h1 + mv_ws.concat + mv_b2 + mv_bs   (B,256,N)
// ---------------------------------------------------------------------------
__global__ void k_mlp_value(const float* __restrict__ kf, const float* __restrict__ qf,
                            const float* __restrict__ h1,
                            const float* __restrict__ w2, const float* __restrict__ b2,
                            const float* __restrict__ ws_, const float* __restrict__ bs,
                            float* __restrict__ value) {
  const int lane = threadIdx.x & 31, hh = lane >> 4, lm = lane & 15;
  const int wvid = threadIdx.x >> 5;
  const int p0 = blockIdx.x * 16;
  const int b  = p0 / N_PTS, n0 = p0 % N_PTS;
  const int m0 = (blockIdx.y * 4 + wvid) * 16;
  v8f acc = zero8();
  for (int c0 = 0; c0 < CIN; c0 += 4) {
    const float* ap = w2 + (size_t)(m0 + lm) * CIN + c0 + 2 * hh;
    v2f a; a.x = ap[0]; a.y = ap[1];
    const float* X = h1 + ((size_t)b * CIN + c0) * N_PTS;
    v2f bb;
    bb.x = X[(size_t)(2 * hh)     * N_PTS + n0 + lm];
    bb.y = X[(size_t)(2 * hh + 1) * N_PTS + n0 + lm];
    acc = wmma_f32(a, bb, acc);
  }
  for (int c0 = 0; c0 < 2 * CIN; c0 += 4) {
    const float* ap = ws_ + (size_t)(m0 + lm) * (2 * CIN) + c0 + 2 * hh;
    v2f a; a.x = ap[0]; a.y = ap[1];
    const float* X = (c0 < CIN) ? (kf + ((size_t)b * CIN + c0) * N_PTS)
                                : (qf + ((size_t)b * CIN + (c0 - CIN)) * N_PTS);
    v2f bb;
    bb.x = X[(size_t)(2 * hh)     * N_PTS + n0 + lm];
    bb.y = X[(size_t)(2 * hh + 1) * N_PTS + n0 + lm];
    acc = wmma_f32(a, bb, acc);
  }
#pragma unroll
  for (int v = 0; v < 8; ++v) {
    int m = m0 + v + 8 * hh;
    value[((size_t)b * CIN + m) * N_PTS + n0 + lm] = acc[v] + b2[m] + bs[m];
  }
}

// ---------------------------------------------------------------------------
// K3: 64-channel projections k/q/v/uf  (selected by blockIdx.z)
// grid(1024,1,4) block(128)
// ---------------------------------------------------------------------------
__global__ void k_proj(const float* __restrict__ kf, const float* __restrict__ qf,
                       const float* __restrict__ value, const float* __restrict__ upf,
                       const float* __restrict__ wk, const float* __restrict__ bk,
                       const float* __restrict__ wq, const float* __restrict__ bq,
                       const float* __restrict__ wv, const float* __restrict__ bv,
                       const float* __restrict__ wu, const float* __restrict__ bu,
                       float* __restrict__ ok, float* __restrict__ oq,
                       float* __restrict__ ov, float* __restrict__ ou) {
  const int lane = threadIdx.x & 31, hh = lane >> 4, lm = lane & 15;
  const int wvid = threadIdx.x >> 5;
  const int p0 = blockIdx.x * 16;
  const int b  = p0 / N_PTS, n0 = p0 % N_PTS;
  const int m0 = wvid * 16;
  const float *W, *bias, *X; float* O;
  switch (blockIdx.z) {
    case 0:  W = wk; bias = bk; X = kf;    O = ok; break;
    case 1:  W = wq; bias = bq; X = qf;    O = oq; break;
    case 2:  W = wv; bias = bv; X = value; O = ov; break;
    default: W = wu; bias = bu; X = upf;   O = ou; break;
  }
  v8f acc = zero8();
  for (int c0 = 0; c0 < CIN; c0 += 4) {
    const float* ap = W + (size_t)(m0 + lm) * CIN + c0 + 2 * hh;
    v2f a; a.x = ap[0]; a.y = ap[1];
    const float* Xp = X + ((size_t)b * CIN + c0) * N_PTS;
    v2f bb;
    bb.x = Xp[(size_t)(2 * hh)     * N_PTS + n0 + lm];
    bb.y = Xp[(size_t)(2 * hh + 1) * N_PTS + n0 + lm];
    acc = wmma_f32(a, bb, acc);
  }
#pragma unroll
  for (int v = 0; v < 8; ++v) {
    int m = m0 + v + 8 * hh;
    O[((size_t)b * DIM + m) * N_PTS + n0 + lm] = acc[v] + bias[m];
  }
}

// ---------------------------------------------------------------------------
// K4: self-KNN (top-20 smallest squared distance, stable ties, includes self)
// ---------------------------------------------------------------------------
__global__ void k_knn(const float* __restrict__ pos, int* __restrict__ idxb) {
  __shared__ float sp[3][256];
  const int tid = threadIdx.x;
  const int gid = blockIdx.x * 256 + tid;
  const int b = gid / N_PTS, n = gid % N_PTS;
  const float* pb = pos + (size_t)b * 3 * N_PTS;
  const float px = pb[n], py = pb[N_PTS + n], pz = pb[2 * N_PTS + n];
  float bd[KNN]; int bi[KNN];
#pragma unroll
  for (int t = 0; t < KNN; ++t) { bd[t] = 3.4e38f; bi[t] = 0; }
  for (int j0 = 0; j0 < N_PTS; j0 += 256) {
    __syncthreads();
    sp[0][tid] = pb[j0 + tid];
    sp[1][tid] = pb[N_PTS + j0 + tid];
    sp[2][tid] = pb[2 * N_PTS + j0 + tid];
    __syncthreads();
    for (int jj = 0; jj < 256; ++jj) {
      float dx = px - sp[0][jj], dy = py - sp[1][jj], dz = pz - sp[2][jj];
      float d = dx * dx + dy * dy + dz * dz;
      int j = j0 + jj;
      if (d < bd[KNN - 1]) {
#pragma unroll
        for (int t = KNN - 1; t >= 0; --t) {
          float pd = (t > 0) ? bd[t - 1] : -1.0f;
          int   pi = (t > 0) ? bi[t - 1] : 0;
          if (d < bd[t]) {
            if (d < pd) { bd[t] = pd; bi[t] = pi; }
            else        { bd[t] = d;  bi[t] = j;  }
          }
        }
      }
    }
  }
#pragma unroll
  for (int t = 0; t < KNN; ++t) idxb[(size_t)gid * KNN + t] = bi[t];
}

// ---------------------------------------------------------------------------
// K6: d_out = repeat(wr.value + br, UP)   (B,128,4096)  -- full overwrite
// grid(1024,2) block(128)
// ---------------------------------------------------------------------------
__global__ void k_iden(const float* __restrict__ value,
                       const float* __restrict__ wr, const float* __restrict__ br,
                       float* __restrict__ out) {
  const int lane = threadIdx.x & 31, hh = lane >> 4, lm = lane & 15;
  const int wvid = threadIdx.x >> 5;
  const int p0 = blockIdx.x * 16;
  const int b  = p0 / N_PTS, n0 = p0 % N_PTS;
  const int m0 = (blockIdx.y * 4 + wvid) * 16;
  v8f acc = zero8();
  for (int c0 = 0; c0 < CIN; c0 += 4) {
    const float* ap = wr + (size_t)(m0 + lm) * CIN + c0 + 2 * hh;
    v2f a; a.x = ap[0]; a.y = ap[1];
    const float* X = value + ((size_t)b * CIN + c0) * N_PTS;
    v2f bb;
    bb.x = X[(size_t)(2 * hh)     * N_PTS + n0 + lm];
    bb.y = X[(size_t)(2 * hh + 1) * N_PTS + n0 + lm];
    acc = wmma_f32(a, bb, acc);
  }
#pragma unroll
  for (int v = 0; v < 8; ++v) {
    int m = m0 + v + 8 * hh;
    float r = acc[v] + br[m];
    size_t base = ((size_t)b * COUT + m) * NUP + 2 * (size_t)(n0 + lm);
    out[base]     = r;
    out[base + 1] = r;
  }
}

// ---------------------------------------------------------------------------
// K5: fused per-point attention: pe MLP, a_in/val, am_w1 GEMM (WMMA),
//     logits GEMM (WMMA), softmax over K, aggregate, we GEMV, += d_out
// grid(16384) block(256) : one workgroup per (b,n)
// ---------------------------------------------------------------------------
__global__ void k_attn(const float* __restrict__ pos,
                       const float* __restrict__ kk, const float* __restrict__ qq,
                       const float* __restrict__ vv, const float* __restrict__ uu,
                       const int*   __restrict__ idxb,
                       const float* __restrict__ pm_w1, const float* __restrict__ pm_b1,
                       const float* __restrict__ pm_g,  const float* __restrict__ pm_be,
                       const float* __restrict__ pm_m,  const float* __restrict__ pm_v,
                       const float* __restrict__ pm_w2, const float* __restrict__ pm_b2,
                       const float* __restrict__ am_w1, const float* __restrict__ am_b1,
                       const float* __restrict__ am_g,  const float* __restrict__ am_be,
                       const float* __restrict__ am_m,  const float* __restrict__ am_v,
                       const float* __restrict__ am_wt, const float* __restrict__ am_bt,
                       const float* __restrict__ we,    const float* __restrict__ be,
                       float* __restrict__ out) {
  __shared__ float s_t[PHID][KNN + 1];
  __shared__ float s_pe[DIM][KNN + 1];
  __shared__ float s_ain[DIM][KNN + 1];
  __shared__ float s_val[DIM][KNN + 1];
  __shared__ float s_h[HID][KNN + 1];
  __shared__ float s_log[COUT][KNN + 1];
  __shared__ float s_prel[3][KNN];
  __shared__ float s_q[DIM], s_u0[DIM];
  __shared__ float s_agg[COUT];
  __shared__ int   s_idx[KNN];

  const int tid  = threadIdx.x;
  const int lane = tid & 31, hh = lane >> 4, lm = lane & 15;
  const int wvid = tid >> 5;
  const int p = blockIdx.x;
  const int b = p / N_PTS, n = p % N_PTS;

  // phase 1: stage idx, q(n), uf(n)
  if (tid < KNN) s_idx[tid] = idxb[(size_t)p * KNN + tid];
  if (tid >= 64  && tid < 128) s_q[tid - 64]  = qq[((size_t)b * DIM + (tid - 64))  * N_PTS + n];
  if (tid >= 128 && tid < 192) s_u0[tid - 128] = uu[((size_t)b * DIM + (tid - 128)) * N_PTS + n];
  __syncthreads();
  if (tid < 3 * KNN) {
    int c = tid / KNN, kq = tid % KNN;
    const float* pc = pos + ((size_t)b * 3 + c) * N_PTS;
    s_prel[c][kq] = pc[n] - pc[s_idx[kq]];
  }
  __syncthreads();

  // phase 2: pe hidden t = relu(bn(pm_w1 . pos_rel + pm_b1))
  for (int it = tid; it < PHID * KNN; it += 256) {
    int ph = it % PHID, kq = it / PHID;
    float t = pm_b1[ph] + pm_w1[ph * 3 + 0] * s_prel[0][kq]
                        + pm_w1[ph * 3 + 1] * s_prel[1][kq]
                        + pm_w1[ph * 3 + 2] * s_prel[2][kq];
    float sc = pm_g[ph] / sqrtf(pm_v[ph] + EPSB);
    t = t * sc + (pm_be[ph] - pm_m[ph] * sc);
    s_t[ph][kq] = fmaxf(t, 0.f);
  }
  __syncthreads();

  // phase 3: pe = pm_w2 . t + pm_b2
  for (int it = tid; it < DIM * KNN; it += 256) {
    int d = it % DIM, kq = it / DIM;
    float acc = pm_b2[d];
    for (int ph = 0; ph < PHID; ++ph) acc += pm_w2[d * PHID + ph] * s_t[ph][kq];
    s_pe[d][kq] = acc;
  }
  __syncthreads();

  // phase 4: a_in = q - k_g + pe + uf_rel ; val = v_g + pe + uf_rel
  for (int it = tid; it < DIM * KNN; it += 256) {
    int d = it % DIM, kq = it / DIM;
    int j = s_idx[kq];
    float ufr = s_u0[d] - uu[((size_t)b * DIM + d) * N_PTS + j];
    float pe  = s_pe[d][kq];
    float kg  = kk[((size_t)b * DIM + d) * N_PTS + j];
    float vg  = vv[((size_t)b * DIM + d) * N_PTS + j];
    s_ain[d][kq] = s_q[d] - kg + pe + ufr;
    s_val[d][kq] = vg + pe + ufr;
  }
  __syncthreads();

  // phase 5 (WMMA): h = relu(bn(am_w1 . a_in + am_b1))   (256 x 20)
  for (int t = 0; t < 4; ++t) {
    int rt = wvid * 2 + (t >> 1);
    int ct = t & 1;
    int m0 = rt * 16;
    int col = ct * 16 + lm;
    v8f acc = zero8();
    for (int c0 = 0; c0 < DIM; c0 += 4) {
      const float* ap = am_w1 + (size_t)(m0 + lm) * DIM + c0 + 2 * hh;
      v2f a; a.x = ap[0]; a.y = ap[1];
      bool ok = col < KNN;
      v2f bb;
      bb.x = ok ? s_ain[c0 + 2 * hh][col]     : 0.f;
      bb.y = ok ? s_ain[c0 + 2 * hh + 1][col] : 0.f;
      acc = wmma_f32(a, bb, acc);
    }
    if (col < KNN) {
#pragma unroll
      for (int v = 0; v < 8; ++v) {
        int m = m0 + v + 8 * hh;
        float sc = am_g[m] / sqrtf(am_v[m] + EPSB);
        float r = (acc[v] + am_b1[m]) * sc + (am_be[m] - am_m[m] * sc);
        s_h[m][col] = fmaxf(r, 0.f);
      }
    }
  }
  __syncthreads();

  // phase 6 (WMMA): logits[m=o*2+r][k] = sum_c h[c][k] * am_wt[c][m] + am_bt[o]
  for (int ct = 0; ct < 2; ++ct) {
    int m0 = wvid * 16;
    int col = ct * 16 + lm;
    v8f acc = zero8();
    for (int c0 = 0; c0 < HID; c0 += 4) {
      v2f a;
      a.x = am_wt[(size_t)(c0 + 2 * hh)     * (DIM * UPF) + m0 + lm];
      a.y = am_wt[(size_t)(c0 + 2 * hh + 1) * (DIM * UPF) + m0 + lm];
      bool ok = col < KNN;
      v2f bb;
      bb.x = ok ? s_h[c0 + 2 * hh][col]     : 0.f;
      bb.y = ok ? s_h[c0 + 2 * hh + 1][col] : 0.f;
      acc = wmma_f32(a, bb, acc);
    }
    if (col < KNN) {
#pragma unroll
      for (int v = 0; v < 8; ++v) {
        int m = m0 + v + 8 * hh;
        s_log[m][col] = acc[v] + am_bt[m >> 1];
      }
    }
  }
  __syncthreads();

  // phase 7: softmax over k + aggregate with val
  if (tid < COUT) {
    int m = tid, o = m >> 1;
    float mx = -3.4e38f;
    for (int kq = 0; kq < KNN; ++kq) mx = fmaxf(mx, s_log[m][kq]);
    float ssum = 0.f, acc = 0.f;
    for (int kq = 0; kq < KNN; ++kq) {
      float e = expf(s_log[m][kq] - mx);
      ssum += e;
      acc  += e * s_val[o][kq];
    }
    s_agg[m] = acc / ssum;
  }
  __syncthreads();

  // phase 8: y = we . agg + be ; out[b, o2, 2n+r] += y
  {
    int o2 = tid >> 1, r = tid & 1;
    float y = be[o2];
    for (int o = 0; o < DIM; ++o) y += we[o2 * DIM + o] * s_agg[o * UPF + r];
    out[((size_t)b * COUT + o2) * NUP + 2 * (size_t)n + r] += y;
  }
}

// ---------------------------------------------------------------------------
extern "C" void kernel_launch(void* const* d_in, const int* in_sizes, int n_in,
                              void* d_out, int out_size, void* d_ws, size_t ws_size,
                              hipStream_t stream) {
  const float* pos   = (const float*)d_in[0];
  const float* kf    = (const float*)d_in[1];
  const float* qf    = (const float*)d_in[2];
  const float* upf   = (const float*)d_in[3];
  const float* mv_w1 = (const float*)d_in[4];
  const float* mv_b1 = (const float*)d_in[5];
  const float* mv_w2 = (const float*)d_in[6];
  const float* mv_b2 = (const float*)d_in[7];
  const float* mv_ws = (const float*)d_in[8];
  const float* mv_bs = (const float*)d_in[9];
  const float* wk    = (const float*)d_in[10];
  const float* bk    = (const float*)d_in[11];
  const float* wq    = (const float*)d_in[12];
  const float* bq    = (const float*)d_in[13];
  const float* wv    = (const float*)d_in[14];
  const float* bv    = (const float*)d_in[15];
  const float* wu    = (const float*)d_in[16];
  const float* bu    = (const float*)d_in[17];
  const float* pm_w1 = (const float*)d_in[18];
  const float* pm_b1 = (const float*)d_in[19];
  const float* pm_g  = (const float*)d_in[20];
  const float* pm_be = (const float*)d_in[21];
  const float* pm_m  = (const float*)d_in[22];
  const float* pm_v  = (const float*)d_in[23];
  const float* pm_w2 = (const float*)d_in[24];
  const float* pm_b2 = (const float*)d_in[25];
  const float* am_w1 = (const float*)d_in[26];
  const float* am_b1 = (const float*)d_in[27];
  const float* am_g  = (const float*)d_in[28];
  const float* am_be = (const float*)d_in[29];
  const float* am_m  = (const float*)d_in[30];
  const float* am_v  = (const float*)d_in[31];
  const float* am_wt = (const float*)d_in[32];
  const float* am_bt = (const float*)d_in[33];
  const float* we    = (const float*)d_in[34];
  const float* be    = (const float*)d_in[35];
  const float* wr    = (const float*)d_in[36];
  const float* br    = (const float*)d_in[37];
  float* out = (float*)d_out;

  const size_t SZ_CN = (size_t)B_SZ * CIN * N_PTS;  // 4,194,304
  const size_t SZ_DN = (size_t)B_SZ * DIM * N_PTS;  // 1,048,576
  float* ws    = (float*)d_ws;
  float* h1    = ws;
  float* value = h1 + SZ_CN;
  float* kkb   = value + SZ_CN;
  float* qqb   = kkb + SZ_DN;
  float* vvb   = qqb + SZ_DN;
  float* uub   = vvb + SZ_DN;
  int*   idxb  = (int*)(uub + SZ_DN);

  const int PT = (B_SZ * N_PTS) / 16;  // 1024 point tiles

  k_mlp_hidden<<<dim3(PT, 4), 128, 0, stream>>>(kf, qf, mv_w1, mv_b1, h1);
  k_mlp_value<<<dim3(PT, 4), 128, 0, stream>>>(kf, qf, h1, mv_w2, mv_b2, mv_ws, mv_bs, value);
  k_proj<<<dim3(PT, 1, 4), 128, 0, stream>>>(kf, qf, value, upf,
                                             wk, bk, wq, bq, wv, bv, wu, bu,
                                             kkb, qqb, vvb, uub);
  k_knn<<<(B_SZ * N_PTS) / 256, 256, 0, stream>>>(pos, idxb);
  k_iden<<<dim3(PT, 2), 128, 0, stream>>>(value, wr, br, out);
  k_attn<<<dim3(B_SZ * N_PTS), 256, 0, stream>>>(pos, kkb, qqb, vvb, uub, idxb,
                                                 pm_w1, pm_b1, pm_g, pm_be, pm_m, pm_v,
                                                 pm_w2, pm_b2,
                                                 am_w1, am_b1, am_g, am_be, am_m, am_v,
                                                 am_wt, am_bt, we, be, out);
}